// HyperDimensionalAdapter_28346784154023
// MI455X (gfx1250) — compile-verified
//
#include <hip/hip_runtime.h>
#include <hip/hip_bf16.h>

typedef __attribute__((ext_vector_type(2))) float v2f;
typedef __attribute__((ext_vector_type(8))) float v8f;

#define HC 16
#define PQ (2048 * 2048)
#define PF (2048 * 4096)

// ---------------------------------------------------------------------------
// Prologue: select s from hypercube, rfft*filter -> irfft -> haar, for both
// heads. Writes h_q to ws[0..15], h_ffn to ws[16..31]. Trivial cost.
// ---------------------------------------------------------------------------
__global__ void hda_prologue(const float* __restrict__ ctx,
                             const float* __restrict__ cube,
                             const float* __restrict__ fw_q,
                             const float* __restrict__ ps_q,
                             const float* __restrict__ fw_f,
                             const float* __restrict__ ps_f,
                             float* __restrict__ hws) {
  if (threadIdx.x != 0 || blockIdx.x != 0) return;

  const int i0 = (int)(ctx[0] * 15.0f);
  const int i1 = (int)(ctx[1] * 15.0f);
  const int i2 = (int)(ctx[2] * 15.0f);
  const float* s = cube + (((size_t)i0 * HC + i1) * HC + i2) * HC;

  float sr[HC];
#pragma unroll
  for (int n = 0; n < HC; ++n) sr[n] = s[n];

  const float PI = 3.14159265358979323846f;
  // rfft: 9 complex bins
  float Sre[9], Sim[9];
  for (int k = 0; k < 9; ++k) {
    float re = 0.0f, im = 0.0f;
    for (int n = 0; n < HC; ++n) {
      const float ang = -2.0f * PI * (float)(k * n) / 16.0f;
      re += sr[n] * cosf(ang);
      im += sr[n] * sinf(ang);
    }
    Sre[k] = re;
    Sim[k] = im;
  }

  for (int head = 0; head < 2; ++head) {
    const float* fw = head ? fw_f : fw_q;
    const float* ps = head ? ps_f : ps_q;
    float Yre[9], Yim[9];
    for (int k = 0; k < 9; ++k) {
      Yre[k] = Sre[k] * fw[k] - Sim[k] * ps[k];
      Yim[k] = Sre[k] * ps[k] + Sim[k] * fw[k];
    }
    // irfft (n=16): imag parts of bins 0 and 8 are dropped (numpy semantics)
    float f[HC];
    for (int n = 0; n < HC; ++n) {
      float acc = Yre[0] + ((n & 1) ? -Yre[8] : Yre[8]);
      for (int k = 1; k < 8; ++k) {
        const float ang = 2.0f * PI * (float)(k * n) / 16.0f;
        acc += 2.0f * (Yre[k] * cosf(ang) - Yim[k] * sinf(ang));
      }
      f[n] = acc * (1.0f / 16.0f);
    }
    // haar
    const float is2 = 0.70710678118654752440f;
    float* ho = hws + head * HC;
    for (int i = 0; i < 8; ++i) {
      ho[i] = (f[2 * i] + f[2 * i + 1]) * is2;
      ho[8 + i] = (f[2 * i] - f[2 * i + 1]) * is2;
    }
  }
}

// ---------------------------------------------------------------------------
// Streaming GEMV (K=16) via V_WMMA_F32_16X16X4_F32.
// Per 16-row tile of W:
//   B operand = W tile chunk (4x16, K-chunk t):  lane n<16 holds
//     W[row n, 4t+0..1], lane n+16 holds W[row n, 4t+2..3]  -> one b64 load.
//   A operand = h chunk broadcast down all 16 rows.
//   D[i,j] (all i equal) = dot(W[row j], h); lanes 0..15 store c[0]+bias.
// Memory-bound: W streamed once, non-temporally, fully coalesced.
// ---------------------------------------------------------------------------
__global__ __launch_bounds__(256) void hda_wmma_gemv(
    const float* __restrict__ W, const float* __restrict__ bias,
    const float* __restrict__ h, float* __restrict__ out, int ntiles) {
  const int lane = threadIdx.x & 31;
  const int wave = blockIdx.x * (blockDim.x >> 5) + (threadIdx.x >> 5);
  const int nwaves = gridDim.x * (blockDim.x >> 5);
  const bool hi = lane >= 16;
  const int col = lane & 15;

  // Uniform h -> per-lane A fragments (constant across tiles).
  float hv[HC];
#pragma unroll
  for (int i = 0; i < HC; ++i) hv[i] = h[i];
  v2f a[4];
#pragma unroll
  for (int t = 0; t < 4; ++t) {
    a[t].x = hi ? hv[4 * t + 2] : hv[4 * t + 0];
    a[t].y = hi ? hv[4 * t + 3] : hv[4 * t + 1];
  }

  const int koff = hi ? 2 : 0;

  for (int tile = wave; tile < ntiles; tile += nwaves) {
    const float* Wt = W + (size_t)tile * (HC * HC);

    // Prefetch this wave's next tile (streams ahead of the WMMA consume).
    if (tile + nwaves < ntiles) {
      __builtin_prefetch(W + (size_t)(tile + nwaves) * (HC * HC) + lane * 8, 0, 0);
    }

    v8f c = {0.f, 0.f, 0.f, 0.f, 0.f, 0.f, 0.f, 0.f};
#pragma unroll
    for (int t = 0; t < 4; ++t) {
      const v2f bfrag = __builtin_nontemporal_load(
          (const v2f*)(Wt + col * HC + 4 * t + koff));
      c = __builtin_amdgcn_wmma_f32_16x16x4_f32(
          /*neg_a=*/false, a[t], /*neg_b=*/false, bfrag,
          /*c_mod=*/(short)0, c, /*reuse_a=*/false, /*reuse_b=*/false);
    }

    if (!hi) {
      const int row = tile * HC + col;
      const float bv = __builtin_nontemporal_load(bias + row);
      __builtin_nontemporal_store(c[0] + bv, out + row);
    }
  }
}

// ---------------------------------------------------------------------------
// Launch
// ---------------------------------------------------------------------------
extern "C" void kernel_launch(void* const* d_in, const int* in_sizes, int n_in,
                              void* d_out, int out_size, void* d_ws,
                              size_t ws_size, hipStream_t stream) {
  const float* ctx  = (const float*)d_in[0];
  const float* cube = (const float*)d_in[1];
  const float* fw_q = (const float*)d_in[2];
  const float* ps_q = (const float*)d_in[3];
  const float* W_q  = (const float*)d_in[4];
  const float* b_q  = (const float*)d_in[5];
  const float* fw_f = (const float*)d_in[6];
  const float* ps_f = (const float*)d_in[7];
  const float* W_f  = (const float*)d_in[8];
  const float* b_f  = (const float*)d_in[9];
  float* out = (float*)d_out;
  float* hws = (float*)d_ws;  // 32 floats: h_q | h_ffn

  hda_prologue<<<1, 32, 0, stream>>>(ctx, cube, fw_q, ps_q, fw_f, ps_f, hws);

  const int ntq = PQ / HC;  // 262144 tiles
  const int ntf = PF / HC;  // 524288 tiles
  const int blocks = 4096;  // 8 waves/block -> 32K waves in flight

  hda_wmma_gemv<<<blocks, 256, 0, stream>>>(W_q, b_q, hws, out, ntq);
  hda_wmma_gemv<<<blocks, 256, 0, stream>>>(W_f, b_f, hws + HC, out + PQ, ntf);
}